// Upsample_61177514164405
// MI455X (gfx1250) — compile-verified
//
#include <hip/hip_runtime.h>

// ---------------------------------------------------------------------------
// Polyphase 2x upsample + depthwise 4x4 FIR (factor=2), bandwidth-bound.
// CDNA5 path: async global->LDS staging (ASYNCcnt), s_wait_asynccnt,
// non-temporal b128 output stores.
// ---------------------------------------------------------------------------

#define AS1 __attribute__((address_space(1)))
#define AS3 __attribute__((address_space(3)))

typedef float v4f __attribute__((ext_vector_type(4)));

__device__ __forceinline__ void async_g2l_f32(const float* gsrc, const float* ldst) {
#if __has_builtin(__builtin_amdgcn_global_load_async_to_lds_b32)
    __builtin_amdgcn_global_load_async_to_lds_b32(
        (AS1 int*)(unsigned long long)gsrc,
        (AS3 int*)(unsigned)(unsigned long long)ldst,
        /*offset=*/0, /*cpol=*/0);
#else
    unsigned l = (unsigned)(unsigned long long)ldst;
    asm volatile("global_load_async_to_lds_b32 %0, %1, off"
                 :: "v"(l), "v"(gsrc) : "memory");
#endif
}

__device__ __forceinline__ void wait_async_zero() {
#if __has_builtin(__builtin_amdgcn_s_wait_asynccnt)
    __builtin_amdgcn_s_wait_asynccnt(0);
#else
    asm volatile("s_wait_asynccnt 0" ::: "memory");
#endif
}

// Problem constants (from reference setup)
constexpr int Hin  = 128, Win  = 128;
constexpr int Hout = 256, Wout = 256;
constexpr int C    = 128;

// Tile: each block covers 16 input rows x 32 input cols -> 32x64 output tile.
constexpr int TILE_X  = 32;          // input cols per block
constexpr int TILE_Y  = 16;          // input rows per block
constexpr int LR      = TILE_Y + 2;  // 18 halo rows
constexpr int LC      = TILE_X + 2;  // 34 halo cols
constexpr int LSTRIDE = 36;          // padded LDS row stride (floats)

__global__ __launch_bounds__(256)
void Upsample_61177514164405_kernel(const float* __restrict__ x,
                                    const float* __restrict__ kern,
                                    float* __restrict__ out)
{
    __shared__ float tile[LR * LSTRIDE];

    const int img = blockIdx.y;                 // b*C + c
    const int ch  = img & (C - 1);
    const int x0  = (blockIdx.x & 3)  * TILE_X; // 128/32 = 4 tiles in x
    const int y0  = (blockIdx.x >> 2) * TILE_Y; // 128/16 = 8 tiles in y

    const float* __restrict__ xin = x + (size_t)img * (size_t)(Hin * Win);
    const float* __restrict__ kv  = kern + (size_t)ch * 16;  // (C,1,4,4)

    const int tid = threadIdx.x;

    // ---- Stage 18x34 halo tile into LDS via async global->LDS copies ------
    for (int e = tid; e < LR * LC; e += 256) {
        const int r   = e / LC;
        const int col = e - r * LC;
        const int gy  = y0 - 1 + r;
        const int gx  = x0 - 1 + col;
        float* ld = &tile[r * LSTRIDE + col];
        if ((unsigned)gy < (unsigned)Hin && (unsigned)gx < (unsigned)Win) {
            async_g2l_f32(xin + gy * Win + gx, ld);
        } else {
            *ld = 0.0f;   // zero padding of the upsampled image
        }
    }
    wait_async_zero();
    __syncthreads();

    // ---- Per-channel 4x4 taps (uniform across block -> scalar loads) ------
    float k[16];
#pragma unroll
    for (int i = 0; i < 16; ++i) k[i] = kv[i];

    // ---- Each thread: two adjacent input quads -> 2 rows x 4 cols output --
    const int tx = tid & 15;   // input col pair index: cols x0+2tx, x0+2tx+1
    const int ty = tid >> 4;   // input row: y0+ty

    const float* s0 = &tile[(ty + 0) * LSTRIDE + 2 * tx]; // row y-1, cols x-1..x+2
    const float* s1 = &tile[(ty + 1) * LSTRIDE + 2 * tx]; // row y
    const float* s2 = &tile[(ty + 2) * LSTRIDE + 2 * tx]; // row y+1

    const float r00 = s0[0], r01 = s0[1], r02 = s0[2], r03 = s0[3];
    const float r10 = s1[0], r11 = s1[1], r12 = s1[2], r13 = s1[3];
    const float r20 = s2[0], r21 = s2[1], r22 = s2[2], r23 = s2[3];

    // Even output row (2y): vertical taps k[0][j] @ row y-1, k[2][j] @ row y
    v4f top, bot;
    top.x = k[0] * r00 + k[2]  * r01 + k[8]  * r10 + k[10] * r11;
    top.y = k[1] * r01 + k[3]  * r02 + k[9]  * r11 + k[11] * r12;
    top.z = k[0] * r01 + k[2]  * r02 + k[8]  * r11 + k[10] * r12;
    top.w = k[1] * r02 + k[3]  * r03 + k[9]  * r12 + k[11] * r13;
    // Odd output row (2y+1): vertical taps k[1][j] @ row y, k[3][j] @ row y+1
    bot.x = k[4] * r10 + k[6]  * r11 + k[12] * r20 + k[14] * r21;
    bot.y = k[5] * r11 + k[7]  * r12 + k[13] * r21 + k[15] * r22;
    bot.z = k[4] * r11 + k[6]  * r12 + k[12] * r21 + k[14] * r22;
    bot.w = k[5] * r12 + k[7]  * r13 + k[13] * r22 + k[15] * r23;

    const int oy = 2 * (y0 + ty);
    const int ox = 2 * x0 + 4 * tx;
    float* __restrict__ obase = out + (size_t)img * (size_t)(Hout * Wout);

    // Output (537 MB) exceeds the 192 MB L2: stream it with non-temporal stores.
    __builtin_nontemporal_store(top, (v4f*)(obase + (size_t)oy * Wout + ox));
    __builtin_nontemporal_store(bot, (v4f*)(obase + (size_t)(oy + 1) * Wout + ox));
}

extern "C" void kernel_launch(void* const* d_in, const int* in_sizes, int n_in,
                              void* d_out, int out_size, void* d_ws, size_t ws_size,
                              hipStream_t stream) {
    const float* x    = (const float*)d_in[0];   // (16,128,128,128) f32
    const float* kern = (const float*)d_in[1];   // (128,1,4,4) f32
    float*       out  = (float*)d_out;           // (16,128,256,256) f32

    dim3 grid(4 * 8, 16 * C, 1);   // 32 tiles per image, 2048 images
    dim3 block(256, 1, 1);
    Upsample_61177514164405_kernel<<<grid, block, 0, stream>>>(x, kern, out);
}